// Baseline_16595753632199
// MI455X (gfx1250) — compile-verified
//
#include <hip/hip_runtime.h>
#include <hip/hip_bf16.h>

#define NCLS   80
#define IMH    256
#define IMW    256
#define NHW    (IMH * IMW)
#define KTOP   100
#define STRIPR 32
#define TILER  36          // 32 rows + 2 halo each side
#define CANDC  3072        // per-class candidate cap (expected ~2600 peaks)
#define NEGF   (-3.402823466e+38f)

typedef unsigned int u32x4 __attribute__((ext_vector_type(4)));
typedef int          i32x4 __attribute__((ext_vector_type(4)));
typedef int          i32x8 __attribute__((ext_vector_type(8)));

// ---------- LDS byte offset of a generic pointer to __shared__ ----------
__device__ __forceinline__ unsigned lds_off_of(const void* p) {
    // generic LDS pointer = {shared_aperture_hi32, lds_byte_offset}
    return (unsigned)(uintptr_t)p;
}

// ---------- CDNA5 async global -> LDS copy (ASYNCcnt path) ----------
__device__ __forceinline__ void async_copy_b128(unsigned lds_byte_off, const void* gptr) {
    asm volatile("global_load_async_to_lds_b128 %0, %1, off"
                 :: "v"(lds_byte_off), "v"(gptr) : "memory");
}
__device__ __forceinline__ void wait_async_zero() {
    asm volatile("s_wait_asynccnt 0" ::: "memory");
}

// ---------- CDNA5 Tensor Data Mover: 2D tile Global -> LDS (TENSORcnt) ----------
// D# layout per CDNA5 ISA.
__device__ __forceinline__ void tdm_load_2d_f32(unsigned lds_byte_off,
                                                const float* gsrc,
                                                int width_elems,   // tile & tensor dim0
                                                int nrows,         // tile & tensor dim1
                                                int row_stride_elems)
{
    const unsigned long long ga = (unsigned long long)(uintptr_t)gsrc;
    u32x4 g0;
    g0[0] = 1u;                                              // count=1, user descriptor
    g0[1] = lds_byte_off;                                    // lds_addr
    g0[2] = (unsigned)(ga & 0xFFFFFFFFu);                    // global_addr[31:0]
    g0[3] = ((unsigned)(ga >> 32) & 0x01FFFFFFu)             // global_addr[56:32]
          | 0x80000000u;                                     // type=2 ("image")
    i32x8 g1;
    g1[0] = (2 << 16);                                       // data_size=2 -> 4 bytes
    g1[1] = (width_elems & 0xFFFF) << 16;                    // tensor_dim0[15:0]
    g1[2] = ((width_elems >> 16) & 0xFFFF)                   // tensor_dim0[31:16]
          | ((nrows & 0xFFFF) << 16);                        // tensor_dim1[15:0]
    g1[3] = ((nrows >> 16) & 0xFFFF)                         // tensor_dim1[31:16]
          | ((width_elems & 0xFFFF) << 16);                  // tile_dim0
    g1[4] = (nrows & 0xFFFF);                                // tile_dim1 (tile_dim2=0)
    g1[5] = row_stride_elems;                                // tensor_dim0_stride[31:0]
    g1[6] = 0;                                               // stride0 hi / stride1 lo
    g1[7] = 0;                                               // stride1 hi
    const i32x4 z4 = {0, 0, 0, 0};                           // groups 2/3: dims>=2 unused
#if __has_include(<hip/amd_detail/amd_gfx1250_TDM.h>)
    const i32x8 z8 = {0, 0, 0, 0, 0, 0, 0, 0};
    __builtin_amdgcn_tensor_load_to_lds(g0, g1, z4, z4, z8, 0);   // therock / clang-23 (6-arg)
#else
    __builtin_amdgcn_tensor_load_to_lds(g0, g1, z4, z4, 0);       // ROCm 7.2 / clang-22 (5-arg)
#endif
}
__device__ __forceinline__ void wait_tensor_zero() {
    __builtin_amdgcn_s_wait_tensorcnt(0);
}

__device__ __forceinline__ float sigmoidf_(float x) { return 1.0f / (1.0f + expf(-x)); }

// =====================================================================
// Stage A: one block per class. 5x5 peak NMS on raw logits (sigmoid is
// monotone), collect peaks, exact radix-select top-100 by sigmoid score.
// Tile staging uses the Tensor Data Mover (one DMA per strip).
// =====================================================================
__global__ __launch_bounds__(256) void stageA_per_class_topk(
    const float* __restrict__ cls_pred,   // [8,80,256,256]; batch 0 used
    float* __restrict__ ws_val,           // [80*100]
    unsigned* __restrict__ ws_idx)        // [80*100]
{
    __shared__ __align__(16) float tile[TILER][IMW];   // 36 KB
    __shared__ float     cand_val[CANDC];              // 12 KB
    __shared__ unsigned  cand_idx[CANDC];              // 12 KB
    __shared__ int       hist[256];                    // 1 KB
    __shared__ int       s_ncand, s_prefix, s_above, s_g, s_e;

    const int tid = threadIdx.x;
    const int c   = blockIdx.x;
    const float* __restrict__ plane = cls_pred + (size_t)c * NHW;  // batch 0

    if (tid == 0) s_ncand = 0;
    __syncthreads();

    for (int strip = 0; strip < IMH / STRIPR; ++strip) {
        const int y0    = strip * STRIPR;
        const int gy_lo = max(0, y0 - 2);
        const int gy_hi = min(IMH - 1, y0 + STRIPR + 1);   // inclusive
        const int nrows = gy_hi - gy_lo + 1;
        const int r0    = gy_lo - (y0 - 2);                // first valid tile row

        // -inf fill for out-of-image halo rows (strips 0 and 7 only)
        for (int t = tid; t < r0 * IMW; t += 256)
            tile[t >> 8][t & (IMW - 1)] = NEGF;
        for (int t = tid; t < (TILER - r0 - nrows) * IMW; t += 256)
            tile[r0 + nrows + (t >> 8)][t & (IMW - 1)] = NEGF;

        // one TDM descriptor moves the whole valid row block into LDS
        if ((tid >> 5) == 0) {   // wave 0 only
            tdm_load_2d_f32(lds_off_of(&tile[r0][0]),
                            plane + (size_t)gy_lo * IMW,
                            IMW, nrows, IMW);
        }
        wait_tensor_zero();      // TENSORcnt==0 (no-op for non-issuing waves)
        __syncthreads();

        // ---- 5x5 peak test (keep iff no strictly-greater neighbor) ----
        for (int t = tid; t < STRIPR * IMW; t += 256) {
            const int ly = t >> 8;          // 0..31
            const int x  = t & (IMW - 1);
            const int tr = ly + 2;          // tile row of center
            const float v = tile[tr][x];
            bool keep = true;
            #pragma unroll
            for (int dy = -2; dy <= 2; ++dy) {
                #pragma unroll
                for (int dx = -2; dx <= 2; ++dx) {
                    const int xx = x + dx;
                    if (xx < 0 || xx >= IMW) continue;
                    keep = keep && (tile[tr + dy][xx] <= v);
                }
            }
            if (keep) {
                const int p = atomicAdd(&s_ncand, 1);
                if (p < CANDC) {
                    cand_val[p] = sigmoidf_(v);
                    cand_idx[p] = (unsigned)((y0 + ly) * IMW + x);
                }
            }
        }
        __syncthreads();        // tile consumed; safe to refill next strip
    }

    const int ncand = min(s_ncand, CANDC);
    float*    outv = ws_val + c * KTOP;
    unsigned* outi = ws_idx + c * KTOP;

    if (ncand <= KTOP) {                    // uniform branch (shared value)
        for (int k = tid; k < KTOP; k += 256) {
            if (k < ncand) { outv[k] = cand_val[k]; outi[k] = cand_idx[k]; }
            else           { outv[k] = 0.0f;        outi[k] = 0u; }
        }
        return;
    }

    // ---- 4-round radix select: exact 100th-value threshold ----
    unsigned prefix = 0; int above = 0;
    for (int round = 0; round < 4; ++round) {
        const int shift = 24 - 8 * round;
        for (int b = tid; b < 256; b += 256) hist[b] = 0;
        __syncthreads();
        for (int i = tid; i < ncand; i += 256) {
            const unsigned key = __float_as_uint(cand_val[i]); // positive floats
            if (round == 0 || (key >> (shift + 8)) == prefix)
                atomicAdd(&hist[(key >> shift) & 0xFF], 1);
        }
        __syncthreads();
        if (tid == 0) {
            int acc = above; int b = 255;
            for (; b > 0; --b) {
                if (acc + hist[b] >= KTOP) break;
                acc += hist[b];
            }
            s_prefix = (int)((prefix << 8) | (unsigned)b);
            s_above  = acc;
        }
        __syncthreads();
        prefix = (unsigned)s_prefix; above = s_above;
    }
    const unsigned T = prefix;              // threshold key; `above` = #(key > T)
    if (tid == 0) { s_g = 0; s_e = 0; }
    __syncthreads();
    const int need_eq = KTOP - above;
    for (int i = tid; i < ncand; i += 256) {
        const unsigned key = __float_as_uint(cand_val[i]);
        if (key > T) {
            const int p = atomicAdd(&s_g, 1);
            outv[p] = cand_val[i]; outi[p] = cand_idx[i];
        } else if (key == T) {
            const int e = atomicAdd(&s_e, 1);
            if (e < need_eq) {
                outv[above + e] = cand_val[i]; outi[above + e] = cand_idx[i];
            }
        }
    }
}

// =====================================================================
// Stage B: global top-100 over the 80*100 pairs (LDS-staged via async
// copies), bitonic-ordered, then box decode.
// =====================================================================
__global__ __launch_bounds__(256) void stageB_global_topk(
    const float* __restrict__ ws_val,     // [8000]
    const unsigned* __restrict__ ws_idx,  // [8000]
    const float* __restrict__ txty,       // [8,2,256,256]; batch 0 used
    const float* __restrict__ twth,       // [8,2,256,256]; batch 0 used
    float* __restrict__ out)              // [600] = bbox[400] | score[100] | cls[100]
{
    __shared__ __align__(16) float sval[NCLS * KTOP];  // 32000 B
    __shared__ int hist[256];
    __shared__ unsigned long long wkey[128];
    __shared__ int s_prefix, s_above, s_g, s_e;

    const int tid = threadIdx.x;
    const int N = NCLS * KTOP;

    // stage all candidate scores into LDS with async copies (ASYNCcnt path)
    for (int t = tid; t < N / 4; t += 256)
        async_copy_b128(lds_off_of(&sval[t * 4]), ws_val + t * 4);
    wait_async_zero();
    __syncthreads();

    // ---- radix select threshold over all 8000 scores ----
    unsigned prefix = 0; int above = 0;
    for (int round = 0; round < 4; ++round) {
        const int shift = 24 - 8 * round;
        for (int b = tid; b < 256; b += 256) hist[b] = 0;
        __syncthreads();
        for (int i = tid; i < N; i += 256) {
            const unsigned key = __float_as_uint(sval[i]);     // >= 0.0f
            if (round == 0 || (key >> (shift + 8)) == prefix)
                atomicAdd(&hist[(key >> shift) & 0xFF], 1);
        }
        __syncthreads();
        if (tid == 0) {
            int acc = above; int b = 255;
            for (; b > 0; --b) {
                if (acc + hist[b] >= KTOP) break;
                acc += hist[b];
            }
            s_prefix = (int)((prefix << 8) | (unsigned)b);
            s_above  = acc;
        }
        __syncthreads();
        prefix = (unsigned)s_prefix; above = s_above;
    }
    const unsigned T = prefix;
    for (int i = tid; i < 128; i += 256) wkey[i] = 0ull;
    if (tid == 0) { s_g = 0; s_e = 0; }
    __syncthreads();

    // ---- collect exactly 100 winners: composite = valbits<<32 | ~slot ----
    const int need_eq = KTOP - above;
    for (int i = tid; i < N; i += 256) {
        const unsigned key = __float_as_uint(sval[i]);
        if (key > T) {
            const int p = atomicAdd(&s_g, 1);
            wkey[p] = ((unsigned long long)key << 32) | (unsigned)(~(unsigned)i);
        } else if (key == T) {
            const int e = atomicAdd(&s_e, 1);
            if (e < need_eq)
                wkey[above + e] = ((unsigned long long)key << 32) | (unsigned)(~(unsigned)i);
        }
    }
    __syncthreads();

    // ---- bitonic sort 128, descending (ties -> lower flat slot first) ----
    for (int ksz = 2; ksz <= 128; ksz <<= 1) {
        for (int j = ksz >> 1; j > 0; j >>= 1) {
            for (int i = tid; i < 128; i += 256) {
                const int ixj = i ^ j;
                if (ixj > i) {
                    const unsigned long long a = wkey[i], b = wkey[ixj];
                    const bool desc = ((i & ksz) == 0);
                    if (desc ? (a < b) : (a > b)) { wkey[i] = b; wkey[ixj] = a; }
                }
            }
            __syncthreads();
        }
    }

    // ---- decode + write all 600 outputs ----
    for (int k = tid; k < KTOP; k += 256) {
        const unsigned long long comp = wkey[k];
        const unsigned slot = ~(unsigned)(comp & 0xFFFFFFFFull);
        const float score = __uint_as_float((unsigned)(comp >> 32));
        const int   cls   = (int)(slot / KTOP);
        const unsigned hw = ws_idx[slot];
        const int gy = (int)(hw >> 8);
        const int gx = (int)(hw & (IMW - 1));

        const float tx = txty[0 * NHW + hw];
        const float ty = txty[1 * NHW + hw];
        const float tw = twth[0 * NHW + hw];
        const float th = twth[1 * NHW + hw];

        // GS = 1: xy = (grid + sigmoid(t)) * STRIDE ; wh = exp(t) * STRIDE
        const float cx = ((float)gx + sigmoidf_(tx)) * 4.0f;
        const float cy = ((float)gy + sigmoidf_(ty)) * 4.0f;
        const float hw2 = expf(tw) * 2.0f;    // wh/2 = exp*4/2
        const float hh2 = expf(th) * 2.0f;
        const float inv = 1.0f / 1024.0f;
        const float x1 = fminf(fmaxf((cx - hw2) * inv, 0.0f), 1.0f);
        const float y1 = fminf(fmaxf((cy - hh2) * inv, 0.0f), 1.0f);
        const float x2 = fminf(fmaxf((cx + hw2) * inv, 0.0f), 1.0f);
        const float y2 = fminf(fmaxf((cy + hh2) * inv, 0.0f), 1.0f);

        out[4 * k + 0] = x1;
        out[4 * k + 1] = y1;
        out[4 * k + 2] = x2;
        out[4 * k + 3] = y2;
        out[4 * KTOP + k] = score;
        out[5 * KTOP + k] = (float)cls;
    }
}

extern "C" void kernel_launch(void* const* d_in, const int* in_sizes, int n_in,
                              void* d_out, int out_size, void* d_ws, size_t ws_size,
                              hipStream_t stream) {
    (void)in_sizes; (void)n_in; (void)out_size; (void)ws_size;
    const float* cls_pred  = (const float*)d_in[0];   // [8,80,256,256]
    const float* txty_pred = (const float*)d_in[1];   // [8,2,256,256]
    const float* twth_pred = (const float*)d_in[2];   // [8,2,256,256]
    float* out = (float*)d_out;

    float*    ws_val = (float*)d_ws;                                  // 8000 floats
    unsigned* ws_idx = (unsigned*)((char*)d_ws + NCLS * KTOP * sizeof(float));

    stageA_per_class_topk<<<NCLS, 256, 0, stream>>>(cls_pred, ws_val, ws_idx);
    stageB_global_topk<<<1, 256, 0, stream>>>(ws_val, ws_idx, txty_pred, twth_pred, out);
}